// StructAttentionLayer_29257317220719
// MI455X (gfx1250) — compile-verified
//
#include <hip/hip_runtime.h>
#include <hip/hip_bf16.h>
#include <cstdint>

#define ATTR_N 50
#define DIM    256
#define LALPHA 0.2f
#define NVEC4  (ATTR_N * DIM / 4)   // 3200 float4 tiles per b

typedef __attribute__((ext_vector_type(2))) float v2f;
typedef __attribute__((ext_vector_type(8))) float v8f;
typedef __attribute__((ext_vector_type(4))) int   v4i;

#if __has_builtin(__builtin_amdgcn_global_load_async_to_lds_b128)
#define USE_ASYNC_LDS 1
#else
#define USE_ASYNC_LDS 0
#endif

__device__ __forceinline__ float wave_sum(float v) {
#pragma unroll
  for (int off = 16; off > 0; off >>= 1) v += __shfl_xor(v, off, 32);
  return v;
}
__device__ __forceinline__ float wave_max(float v) {
#pragma unroll
  for (int off = 16; off > 0; off >>= 1) v = fmaxf(v, __shfl_xor(v, off, 32));
  return v;
}

__global__ __launch_bounds__(256, 1) void struct_attn_kernel(
    const float* __restrict__ attrs, const float* __restrict__ ent,
    const float* __restrict__ a, float* __restrict__ out) {
  __shared__ float s_attrs[ATTR_N * DIM];  // 51200 B: the per-b tile, read once from HBM
  __shared__ float s_a_attr[DIM];          // a[:256]
  __shared__ float s_att[64];              // e then attention; rows 50..63 zero-padded
  __shared__ float s_red[8];               // per-wave partials of ent[b,:]·a_ent

  const int tid  = threadIdx.x;
  const int lane = tid & 31;
  const int wid  = tid >> 5;
  const int b    = blockIdx.x;

  const float* __restrict__ gsrc = attrs + (size_t)b * (ATTR_N * DIM);

  // ---- Phase 0: async-stage attrs[b] (50x256 f32) into LDS ----
#if USE_ASYNC_LDS
#pragma unroll
  for (int i = 0; i < 13; ++i) {
    const int idx = i * 256 + tid;
    if (idx < NVEC4) {
      __builtin_amdgcn_global_load_async_to_lds_b128(
          (__attribute__((address_space(1))) v4i*)(uintptr_t)(gsrc + idx * 4),
          (__attribute__((address_space(3))) v4i*)(uint32_t)(uintptr_t)(&s_attrs[idx * 4]),
          0, 0);
    }
  }
#else
#pragma unroll
  for (int i = 0; i < 13; ++i) {
    const int idx = i * 256 + tid;
    if (idx < NVEC4)
      reinterpret_cast<float4*>(s_attrs)[idx] =
          reinterpret_cast<const float4*>(gsrc)[idx];
  }
#endif

  // Overlap: a_attr to LDS, and entity-dot partial (256 dims over 256 threads)
  s_a_attr[tid] = a[tid];
  {
    float p = ent[(size_t)b * DIM + tid] * a[DIM + tid];
    p = wave_sum(p);                    // deterministic fixed-order tree
    if (lane == 0) s_red[wid] = p;
  }

#if USE_ASYNC_LDS
#if __has_builtin(__builtin_amdgcn_s_wait_asynccnt)
  __builtin_amdgcn_s_wait_asynccnt(0);
#else
  asm volatile("s_wait_asynccnt 0" ::: "memory");
#endif
#endif
  __syncthreads();

  // ---- Phase 1: e[r] = attrs[b,r,:]·a_attr + entdot, leaky_relu ----
  for (int r = wid; r < ATTR_N; r += 8) {
    float acc = 0.f;
#pragma unroll
    for (int kk = 0; kk < 8; ++kk) {
      const int k = lane + kk * 32;
      acc = fmaf(s_attrs[r * DIM + k], s_a_attr[k], acc);
    }
    acc = wave_sum(acc);
    if (lane == 0) {
      const float entd = ((s_red[0] + s_red[1]) + (s_red[2] + s_red[3])) +
                         ((s_red[4] + s_red[5]) + (s_red[6] + s_red[7]));
      float e = acc + entd;
      s_att[r] = (e > 0.f) ? e : LALPHA * e;
    }
  }
  __syncthreads();

  // ---- Softmax over 50 rows (wave 0), scaled by ATTR_N; pad 50..63 with 0 ----
  if (wid == 0) {
    const float e0 = s_att[lane];                                  // rows 0..31
    const bool  v1 = (lane + 32) < ATTR_N;                          // rows 32..49
    const float e1 = v1 ? s_att[lane + 32] : -3.0e38f;
    const float m  = wave_max(fmaxf(e0, e1));
    const float p0 = expf(e0 - m);
    const float p1 = v1 ? expf(e1 - m) : 0.f;
    const float s  = wave_sum(p0 + p1);
    const float sc = (float)ATTR_N / s;
    s_att[lane]      = p0 * sc;
    s_att[lane + 32] = p1 * sc;   // rows >= 50 get exact 0 (padding for WMMA K)
  }
  __syncthreads();

  // ---- Phase 2: out[b,:] = att(1x50) x attrs(50x256) via V_WMMA_F32_16X16X4_F32 ----
  // A (16x4): att broadcast into all 16 M-rows. Lanes 0-15 hold K0,K1; lanes 16-31 K2,K3.
  // B (4x16): VGPR v holds row v (lanes 0-15) / row v+2 (lanes 16-31), N = lane%16.
  // D (16x16): all rows identical; row 0 lives in acc[0] of lanes 0-15 (N = lane).
  const int khalf = (lane < 16) ? 0 : 2;
  const int nlane = lane & 15;
#pragma unroll
  for (int cc = 0; cc < 2; ++cc) {
    const int n0  = (wid * 2 + cc) * 16;
    const int col = n0 + nlane;
    v8f acc = {};
#pragma unroll
    for (int kk = 0; kk < 13; ++kk) {          // K = 0..51 (rows >= 50 are zeros)
      const int k0 = kk * 4 + khalf;
      v2f av, bv;
      av.x = s_att[k0];
      av.y = s_att[k0 + 1];
      bv.x = (k0     < ATTR_N) ? s_attrs[k0 * DIM + col]       : 0.f;
      bv.y = (k0 + 1 < ATTR_N) ? s_attrs[(k0 + 1) * DIM + col] : 0.f;
      acc = __builtin_amdgcn_wmma_f32_16x16x4_f32(
          false, av, false, bv, (short)0, acc, false, false);
    }
    if (lane < 16) out[(size_t)b * DIM + col] = acc[0];
  }
}

extern "C" void kernel_launch(void* const* d_in, const int* in_sizes, int n_in,
                              void* d_out, int out_size, void* d_ws, size_t ws_size,
                              hipStream_t stream) {
  (void)n_in; (void)d_ws; (void)ws_size; (void)out_size;
  const float* attrs = (const float*)d_in[0];   // (B, 50, 256) f32
  const float* entv  = (const float*)d_in[1];   // (B, 256) f32
  const float* avec  = (const float*)d_in[2];   // (512, 1) f32
  float* out = (float*)d_out;                   // (B, 256) f32
  const int B = in_sizes[1] / DIM;              // 16384
  struct_attn_kernel<<<dim3(B), dim3(256), 0, stream>>>(attrs, entv, avec, out);
}